// DETM_52793738002551
// MI455X (gfx1250) — compile-verified
//
#include <hip/hip_runtime.h>
#include <hip/hip_bf16.h>
#include <math.h>

// ---------------------------------------------------------------------------
// DETM forward (eval) for MI455X / gfx1250, fp32 WMMA (V_WMMA_F32_16X16X4_F32)
// ---------------------------------------------------------------------------

#define Bb 128
#define Vv 30000
#define Tt 50
#define Kk 50
#define HE 256
#define TH 800
#define Ee 300
#define Ll 3

typedef __attribute__((ext_vector_type(2))) float v2f;
typedef __attribute__((ext_vector_type(8))) float v8f;

#define LOGD (-5.2983173665480363f) /* log(0.005) */

#define WMMA4(a, b, c) __builtin_amdgcn_wmma_f32_16x16x4_f32( \
    false, (a), false, (b), (short)0, (c), false, false)

// ---------------------------------------------------------------------------
// Generic WMMA GEMM:  C[M,N] = act( A[M,Kd] * W[N,Kd]^T + bias[N] )
// A row-major lda=Kd, W row-major ldw=Kd, C row-major ldc=N.
// One wave per 32x64 macro-tile (2 M-tiles x 4 N-tiles, 8 f32 accumulators).
// K-loop in steps of 4 using V_WMMA_F32_16X16X4_F32.
// Fragment layouts per CDNA5 ISA 7.12.2:
//   A:  lane = 16*k_half + m ; vgpr j holds A[m][kbase + 2*k_half + j]
//   B:  lane = 16*k_half + n ; vgpr j holds B[kbase + 2*k_half + j][n]
//   D:  lane = 16*row_half + n ; vgpr i holds D[8*row_half + i][n]
// Edge tiles: loads row-clamped (garbage results), stores guarded, so EXEC
// stays all-ones at every WMMA (ISA requirement).
// act: 0 = none, 1 = tanh
// ---------------------------------------------------------------------------
__global__ __launch_bounds__(32) void wmma_gemm_atb(
    const float* __restrict__ A, const float* __restrict__ W,
    const float* __restrict__ bias, float* __restrict__ C,
    int M, int N, int Kd, int act)
{
    const int lane = threadIdx.x & 31;
    const int l16  = lane & 15;
    const int half = lane >> 4;          // 0 or 1
    const int m0 = blockIdx.y * 32;
    const int n0 = blockIdx.x * 64;

    // row-clamped fragment pointers (invalid rows produce discarded results)
    int ar0 = m0 + l16;        if (ar0 >= M) ar0 = M - 1;
    int ar1 = m0 + 16 + l16;   if (ar1 >= M) ar1 = M - 1;
    const float* ap0 = A + (size_t)ar0 * (size_t)Kd + 2 * half;
    const float* ap1 = A + (size_t)ar1 * (size_t)Kd + 2 * half;

    const float* wp[4];
#pragma unroll
    for (int j = 0; j < 4; ++j) {
        int wr = n0 + 16 * j + l16;  if (wr >= N) wr = N - 1;
        wp[j] = W + (size_t)wr * (size_t)Kd + 2 * half;
    }

    v8f acc[2][4];
#pragma unroll
    for (int mi = 0; mi < 2; ++mi)
#pragma unroll
        for (int nj = 0; nj < 4; ++nj)
            acc[mi][nj] = (v8f){0.f, 0.f, 0.f, 0.f, 0.f, 0.f, 0.f, 0.f};

    const int k4 = Kd & ~3;
    for (int k = 0; k < k4; k += 4) {
        // prefetch ~2KB ahead on the streamed operands (speculative, safe OOB)
        __builtin_prefetch(ap0 + k + 512, 0, 1);
        __builtin_prefetch(wp[0] + k + 512, 0, 1);

        v2f a0, a1, b[4];
        a0.x = ap0[k]; a0.y = ap0[k + 1];
        a1.x = ap1[k]; a1.y = ap1[k + 1];
#pragma unroll
        for (int j = 0; j < 4; ++j) { b[j].x = wp[j][k]; b[j].y = wp[j][k + 1]; }

#pragma unroll
        for (int nj = 0; nj < 4; ++nj) {
            acc[0][nj] = WMMA4(a0, b[nj], acc[0][nj]);
            acc[1][nj] = WMMA4(a1, b[nj], acc[1][nj]);
        }
    }
    if (k4 < Kd) {                        // K tail (e.g. Kd = 30050)
        const int ka = k4 + 2 * half;
        v2f a0, a1, b[4];
        a0.x = (ka     < Kd) ? ap0[k4]     : 0.f;
        a0.y = (ka + 1 < Kd) ? ap0[k4 + 1] : 0.f;
        a1.x = (ka     < Kd) ? ap1[k4]     : 0.f;
        a1.y = (ka + 1 < Kd) ? ap1[k4 + 1] : 0.f;
#pragma unroll
        for (int j = 0; j < 4; ++j) {
            b[j].x = (ka     < Kd) ? wp[j][k4]     : 0.f;
            b[j].y = (ka + 1 < Kd) ? wp[j][k4 + 1] : 0.f;
        }
#pragma unroll
        for (int nj = 0; nj < 4; ++nj) {
            acc[0][nj] = WMMA4(a0, b[nj], acc[0][nj]);
            acc[1][nj] = WMMA4(a1, b[nj], acc[1][nj]);
        }
    }

    // store 32x64 macro-tile
#pragma unroll
    for (int nj = 0; nj < 4; ++nj) {
        const int ccol = n0 + 16 * nj + l16;
        if (ccol < N) {
            const float bv = bias ? bias[ccol] : 0.0f;
#pragma unroll
            for (int mi = 0; mi < 2; ++mi) {
                const int rbase = m0 + 16 * mi + 8 * half;
#pragma unroll
                for (int i = 0; i < 8; ++i) {
                    const int crow = rbase + i;
                    if (crow < M) {
                        float v = acc[mi][nj][i] + bv;
                        if (act == 1) v = tanhf(v);
                        C[(size_t)crow * (size_t)N + ccol] = v;
                    }
                }
            }
        }
    }
}

// ---------------------------------------------------------------------------
// Per-row sum of X -> reciprocal (for nb = X / rowsum)
// ---------------------------------------------------------------------------
__global__ void rowsum_inv(const float* __restrict__ X, float* __restrict__ inv, int ncols)
{
    __shared__ float s[256];
    const int b = blockIdx.x;
    float acc = 0.f;
    for (int v = threadIdx.x; v < ncols; v += 256)
        acc += X[(size_t)b * ncols + v];
    s[threadIdx.x] = acc; __syncthreads();
    for (int off = 128; off > 0; off >>= 1) {
        if (threadIdx.x < off) s[threadIdx.x] += s[threadIdx.x + off];
        __syncthreads();
    }
    if (threadIdx.x == 0) inv[b] = 1.0f / s[0];
}

// ---------------------------------------------------------------------------
// 3-layer LSTM over T=50 steps, single workgroup (1024 threads), state in LDS
// ---------------------------------------------------------------------------
__global__ __launch_bounds__(1024) void lstm_kernel(
    const float* __restrict__ rnn_in, const float* __restrict__ Wih,
    const float* __restrict__ Whh, const float* __restrict__ bih,
    const float* __restrict__ bhh, float* __restrict__ rnn_out)
{
    __shared__ float xseq[Tt * HE];   // 50*256 floats
    __shared__ float g[4 * HE];       // 1024 gates
    __shared__ float h[HE], c[HE];
    const int tid = threadIdx.x;

    for (int i = tid; i < Tt * HE; i += 1024) xseq[i] = rnn_in[i];

    for (int l = 0; l < Ll; ++l) {
        __syncthreads();
        if (tid < HE) { h[tid] = 0.f; c[tid] = 0.f; }
        const float* Wi = Wih + (size_t)l * 4 * HE * HE + (size_t)tid * HE;
        const float* Wh = Whh + (size_t)l * 4 * HE * HE + (size_t)tid * HE;
        const float  bs = bih[l * 4 * HE + tid] + bhh[l * 4 * HE + tid];

        for (int t = 0; t < Tt; ++t) {
            __syncthreads();
            const float* xt = &xseq[t * HE];
            float acc = bs;
            for (int j = 0; j < HE; ++j) acc += Wi[j] * xt[j];
            for (int j = 0; j < HE; ++j) acc += Wh[j] * h[j];
            g[tid] = acc;
            __syncthreads();
            if (tid < HE) {
                const float iv = 1.0f / (1.0f + expf(-g[tid]));
                const float fv = 1.0f / (1.0f + expf(-g[HE + tid]));
                const float gv = tanhf(g[2 * HE + tid]);
                const float ov = 1.0f / (1.0f + expf(-g[3 * HE + tid]));
                const float cn = fv * c[tid] + iv * gv;
                c[tid] = cn;
                const float hn = ov * tanhf(cn);
                h[tid] = hn;
                xseq[t * HE + tid] = hn;   // output of this layer = input of next
            }
        }
        __syncthreads();
    }
    for (int i = tid; i < Tt * HE; i += 1024) rnn_out[i] = xseq[i];
}

// ---------------------------------------------------------------------------
// eta scan: sequential Gaussian recurrence + KL(eta)
// ---------------------------------------------------------------------------
__global__ __launch_bounds__(64) void eta_scan(
    const float* __restrict__ rnn, const float* __restrict__ Wmu,
    const float* __restrict__ bmu, const float* __restrict__ Wls,
    const float* __restrict__ bls, float* __restrict__ etas,
    float* __restrict__ kl_out)
{
    __shared__ float out_t[HE], eprev[Kk], nmu[Kk], nls[Kk], kl[1];
    const int tid = threadIdx.x;
    if (tid < Kk) eprev[tid] = 0.f;
    if (tid == 0) kl[0] = 0.f;

    for (int t = 0; t < Tt; ++t) {
        __syncthreads();
        for (int j = tid; j < HE; j += 64) out_t[j] = rnn[t * HE + j];
        __syncthreads();
        if (tid < Kk) {
            const float* wm = Wmu + tid * (HE + Kk);
            const float* wl = Wls + tid * (HE + Kk);
            float m = bmu[tid], l = bls[tid];
            for (int j = 0; j < HE; ++j) { m += wm[j] * out_t[j]; l += wl[j] * out_t[j]; }
            for (int j = 0; j < Kk; ++j) { m += wm[HE + j] * eprev[j]; l += wl[HE + j] * eprev[j]; }
            nmu[tid] = m; nls[tid] = l;
        }
        __syncthreads();
        if (tid == 0) {
            const float pls = (t == 0) ? 0.0f : LOGD;
            const float denom = expf(pls) + 1e-6f;
            float s = 0.f;
            for (int k = 0; k < Kk; ++k) {
                const float d = nmu[k] - eprev[k];
                s += 0.5f * ((expf(nls[k]) + d * d) / denom - 1.0f + pls - nls[k]);
            }
            kl[0] += s;
        }
        __syncthreads();
        if (tid < Kk) { eprev[tid] = nmu[tid]; etas[t * Kk + tid] = nmu[tid]; }
    }
    __syncthreads();
    if (tid == 0) kl_out[0] = kl[0];
}

// ---------------------------------------------------------------------------
// Build concatenated theta-net input: At[b, 0:V]=X*inv, At[b, V:V+K]=etas[ts[b]]
// ---------------------------------------------------------------------------
__global__ void build_atheta(const float* __restrict__ X, const float* __restrict__ inv,
                             const float* __restrict__ etas, const int* __restrict__ ts,
                             float* __restrict__ At)
{
    const int b = blockIdx.x;
    const int j = blockIdx.y * 256 + threadIdx.x;
    if (j < Vv + Kk) {
        float v;
        if (j < Vv) v = X[(size_t)b * Vv + j] * inv[b];
        else        v = etas[ts[b] * Kk + (j - Vv)];
        At[(size_t)b * (Vv + Kk) + j] = v;
    }
}

// ---------------------------------------------------------------------------
// theta softmax (into d_out) + KL(theta) per-row partials
// ---------------------------------------------------------------------------
__global__ __launch_bounds__(64) void theta_klt(
    const float* __restrict__ mu_t, const float* __restrict__ ls_t,
    const float* __restrict__ etas, const int* __restrict__ ts,
    float* __restrict__ theta_out, float* __restrict__ klt_part)
{
    __shared__ float mu[Kk], ex[Kk], kt[64], red[2];
    const int b = blockIdx.x, tid = threadIdx.x;
    float klv = 0.f;
    if (tid < Kk) {
        const float m = mu_t[b * Kk + tid];
        const float l = ls_t[b * Kk + tid];
        const float e = etas[ts[b] * Kk + tid];
        mu[tid] = m;
        const float d = m - e;
        klv = 0.5f * ((expf(l) + d * d) / (1.0f + 1e-6f) - 1.0f - l);
    }
    kt[tid] = klv; __syncthreads();
    if (tid == 0) {
        float mx = mu[0];
        for (int k = 1; k < Kk; ++k) mx = fmaxf(mx, mu[k]);
        red[0] = mx;
        float s = 0.f;
        for (int k = 0; k < 64; ++k) s += kt[k];
        klt_part[b] = s;
    }
    __syncthreads();
    if (tid < Kk) ex[tid] = expf(mu[tid] - red[0]);
    __syncthreads();
    if (tid == 0) { float s = 0.f; for (int k = 0; k < Kk; ++k) s += ex[k]; red[1] = s; }
    __syncthreads();
    if (tid < Kk) theta_out[b * Kk + tid] = ex[tid] / red[1];
}

// ---------------------------------------------------------------------------
// alphas[t,k,e] = mu_q_alpha[k,t,e]  (transpose for the beta GEMM)
// ---------------------------------------------------------------------------
__global__ void transpose_alpha(const float* __restrict__ mqa, float* __restrict__ at)
{
    const int idx = blockIdx.x * 256 + threadIdx.x;
    if (idx < Tt * Kk * Ee) {
        const int e = idx % Ee;
        const int m = idx / Ee;      // m = t*K + k
        const int t = m / Kk, k = m % Kk;
        at[idx] = mqa[((size_t)k * Tt + t) * Ee + e];
    }
}

// ---------------------------------------------------------------------------
// KL(alpha) per-(t,k) partials
// ---------------------------------------------------------------------------
__global__ void kl_alpha_part(const float* __restrict__ mqa, const float* __restrict__ lqa,
                              float* __restrict__ part)
{
    __shared__ float s[256];
    const int m = blockIdx.x;          // t*K + k
    const int t = m / Kk, k = m % Kk;
    const float* a  = mqa + ((size_t)k * Tt + t) * Ee;
    const float* ql = lqa + ((size_t)k * Tt + t) * Ee;
    const float* pm = (t == 0) ? nullptr : mqa + ((size_t)k * Tt + (t - 1)) * Ee;
    const float pls = (t == 0) ? 0.0f : LOGD;
    const float denom = expf(pls) + 1e-6f;
    float acc = 0.f;
    for (int e = threadIdx.x; e < Ee; e += 256) {
        const float d = a[e] - (pm ? pm[e] : 0.0f);
        acc += 0.5f * ((expf(ql[e]) + d * d) / denom - 1.0f + pls - ql[e]);
    }
    s[threadIdx.x] = acc; __syncthreads();
    for (int off = 128; off > 0; off >>= 1) {
        if (threadIdx.x < off) s[threadIdx.x] += s[threadIdx.x + off];
        __syncthreads();
    }
    if (threadIdx.x == 0) part[m] = s[0];
}

// ---------------------------------------------------------------------------
// In-place row softmax (beta rows, ncols=30000)
// ---------------------------------------------------------------------------
__global__ void softmax_rows(float* __restrict__ Mrow, int ncols)
{
    __shared__ float s[256];
    float* p = Mrow + (size_t)blockIdx.x * ncols;
    const int tid = threadIdx.x;
    float mx = -3.4e38f;
    for (int v = tid; v < ncols; v += 256) mx = fmaxf(mx, p[v]);
    s[tid] = mx; __syncthreads();
    for (int off = 128; off > 0; off >>= 1) {
        if (tid < off) s[tid] = fmaxf(s[tid], s[tid + off]);
        __syncthreads();
    }
    mx = s[0]; __syncthreads();
    float sum = 0.f;
    for (int v = tid; v < ncols; v += 256) sum += expf(p[v] - mx);
    s[tid] = sum; __syncthreads();
    for (int off = 128; off > 0; off >>= 1) {
        if (tid < off) s[tid] += s[tid + off];
        __syncthreads();
    }
    const float inv = 1.0f / s[0];
    __syncthreads();
    for (int v = tid; v < ncols; v += 256) p[v] = expf(p[v] - mx) * inv;
}

// ---------------------------------------------------------------------------
// px = theta · beta[ts[b]];  lnpx = log(px+1e-6);  nll partials
// ---------------------------------------------------------------------------
__global__ void px_kernel(const float* __restrict__ theta, const float* __restrict__ beta,
                          const float* __restrict__ X, const int* __restrict__ ts,
                          float* __restrict__ lnpx, float* __restrict__ nll_part)
{
    __shared__ float th[Kk];
    __shared__ float s[256];
    const int b = blockIdx.x, vb = blockIdx.y, tid = threadIdx.x;
    if (tid < Kk) th[tid] = theta[b * Kk + tid];
    __syncthreads();
    const int v = vb * 256 + tid;
    float contrib = 0.f;
    if (v < Vv) {
        const float* brow = beta + (size_t)(ts[b] * Kk) * Vv + v;
        float p = 0.f;
        for (int k = 0; k < Kk; ++k) p += th[k] * brow[(size_t)k * Vv];
        const float ln = logf(p + 1e-6f);
        lnpx[(size_t)b * Vv + v] = ln;
        contrib = -ln * X[(size_t)b * Vv + v];
    }
    s[tid] = contrib; __syncthreads();
    for (int off = 128; off > 0; off >>= 1) {
        if (tid < off) s[tid] += s[tid + off];
        __syncthreads();
    }
    if (tid == 0) nll_part[b * gridDim.y + vb] = s[0];
}

// ---------------------------------------------------------------------------
// Deterministic single-block reduction: dst = scale * sum(src[0..n))
// ---------------------------------------------------------------------------
__global__ void reduce_sum(const float* __restrict__ src, int n, float scale,
                           float* __restrict__ dst)
{
    __shared__ float s[256];
    float acc = 0.f;
    for (int i = threadIdx.x; i < n; i += 256) acc += src[i];
    s[threadIdx.x] = acc; __syncthreads();
    for (int off = 128; off > 0; off >>= 1) {
        if (threadIdx.x < off) s[threadIdx.x] += s[threadIdx.x + off];
        __syncthreads();
    }
    if (threadIdx.x == 0) dst[0] = s[0] * scale;
}

// ---------------------------------------------------------------------------
// Launch
// ---------------------------------------------------------------------------
extern "C" void kernel_launch(void* const* d_in, const int* in_sizes, int n_in,
                              void* d_out, int out_size, void* d_ws, size_t ws_size,
                              hipStream_t stream)
{
    const float* X         = (const float*)d_in[0];
    const int*   ts        = (const int*)  d_in[1];
    const float* X_mean    = (const float*)d_in[2];
    const float* W_eta_map = (const float*)d_in[3];
    const float* b_eta_map = (const float*)d_in[4];
    const float* lstm_Wih  = (const float*)d_in[5];
    const float* lstm_Whh  = (const float*)d_in[6];
    const float* lstm_bih  = (const float*)d_in[7];
    const float* lstm_bhh  = (const float*)d_in[8];
    const float* W_qeta_mu = (const float*)d_in[9];
    const float* b_qeta_mu = (const float*)d_in[10];
    const float* W_qeta_ls = (const float*)d_in[11];
    const float* b_qeta_ls = (const float*)d_in[12];
    const float* W1        = (const float*)d_in[13];
    const float* b1        = (const float*)d_in[14];
    const float* W2        = (const float*)d_in[15];
    const float* b2        = (const float*)d_in[16];
    const float* W_qt_mu   = (const float*)d_in[17];
    const float* b_qt_mu   = (const float*)d_in[18];
    const float* W_qt_ls   = (const float*)d_in[19];
    const float* b_qt_ls   = (const float*)d_in[20];
    const float* mu_q_a    = (const float*)d_in[21];
    const float* ls_q_a    = (const float*)d_in[22];
    const float* emb       = (const float*)d_in[23];

    float* out = (float*)d_out;
    float* ws  = (float*)d_ws;

    // d_out layout (floats): theta, lnpx, beta, nll, kl_alpha, kl_eta, kl_theta
    const size_t OFF_THETA = 0;
    const size_t OFF_LNPX  = OFF_THETA + (size_t)Bb * Kk;          // 6400
    const size_t OFF_BETA  = OFF_LNPX  + (size_t)Bb * Vv;          // 3,846,400
    const size_t OFF_NLL   = OFF_BETA  + (size_t)Tt * Kk * Vv;     // 78,846,400
    const size_t OFF_KLA   = OFF_NLL + 1;
    const size_t OFF_KLE   = OFF_NLL + 2;
    const size_t OFF_KLT   = OFF_NLL + 3;

    // workspace layout (floats) -- total ~19.4 MB
    const size_t WS_INV  = 0;                         // 128
    const size_t WS_RNN  = WS_INV  + Bb;              // 12800
    const size_t WS_ETAS = WS_RNN  + (size_t)Tt * HE; // 2500
    const size_t WS_MU   = WS_ETAS + (size_t)Tt * Kk; // 6400
    const size_t WS_LS   = WS_MU   + (size_t)Bb * Kk; // 6400
    const size_t WS_KLT  = WS_LS   + (size_t)Bb * Kk; // 128
    const size_t WS_KLA  = WS_KLT  + Bb;              // 2500
    const size_t WS_NLL  = WS_KLA  + (size_t)Tt * Kk; // 128*118
    const size_t WS_H1   = WS_NLL  + (size_t)Bb * 118;        // 102400
    const size_t WS_H2   = WS_H1   + (size_t)Bb * TH;         // 102400
    const size_t WS_ALPH = WS_H2   + (size_t)Bb * TH;         // 750000
    const size_t WS_AT   = WS_ALPH + (size_t)Tt * Kk * Ee;    // 3,846,400

    const int NVB = (Vv + 255) / 256; // 118

    // macro-tile grid helper: 64 cols x 32 rows per wave
#define GEMM_GRID(M_, N_) dim3(((N_) + 63) / 64, ((M_) + 31) / 32)

    // 1) row sums of X
    rowsum_inv<<<Bb, 256, 0, stream>>>(X, ws + WS_INV, Vv);

    // 2) rnn_inp = X_mean @ W_eta_map^T + b   (M=50, N=256, K=30000)
    wmma_gemm_atb<<<GEMM_GRID(Tt, HE), 32, 0, stream>>>(
        X_mean, W_eta_map, b_eta_map, ws + WS_RNN, Tt, HE, Vv, 0);

    // 3) 3-layer LSTM (in-place on rnn buffer)
    lstm_kernel<<<1, 1024, 0, stream>>>(ws + WS_RNN, lstm_Wih, lstm_Whh,
                                        lstm_bih, lstm_bhh, ws + WS_RNN);

    // 4) eta scan -> etas + kl_eta
    eta_scan<<<1, 64, 0, stream>>>(ws + WS_RNN, W_qeta_mu, b_qeta_mu,
                                   W_qeta_ls, b_qeta_ls, ws + WS_ETAS, out + OFF_KLE);

    // 5) concat input [nb, eta_td]
    build_atheta<<<dim3(Bb, (Vv + Kk + 255) / 256), 256, 0, stream>>>(
        X, ws + WS_INV, ws + WS_ETAS, ts, ws + WS_AT);

    // 6) h1 = tanh(At @ W1^T + b1)   (M=128, N=800, K=30050)
    wmma_gemm_atb<<<GEMM_GRID(Bb, TH), 32, 0, stream>>>(
        ws + WS_AT, W1, b1, ws + WS_H1, Bb, TH, Vv + Kk, 1);

    // 7) h2 = tanh(h1 @ W2^T + b2)   (M=128, N=800, K=800)
    wmma_gemm_atb<<<GEMM_GRID(Bb, TH), 32, 0, stream>>>(
        ws + WS_H1, W2, b2, ws + WS_H2, Bb, TH, TH, 1);

    // 8) mu_t / ls_t heads           (M=128, N=50, K=800)
    wmma_gemm_atb<<<GEMM_GRID(Bb, Kk), 32, 0, stream>>>(
        ws + WS_H2, W_qt_mu, b_qt_mu, ws + WS_MU, Bb, Kk, TH, 0);
    wmma_gemm_atb<<<GEMM_GRID(Bb, Kk), 32, 0, stream>>>(
        ws + WS_H2, W_qt_ls, b_qt_ls, ws + WS_LS, Bb, Kk, TH, 0);

    // 9) theta softmax + kl_theta partials, then reduce
    theta_klt<<<Bb, 64, 0, stream>>>(ws + WS_MU, ws + WS_LS, ws + WS_ETAS, ts,
                                     out + OFF_THETA, ws + WS_KLT);
    reduce_sum<<<1, 256, 0, stream>>>(ws + WS_KLT, Bb, 1.0f, out + OFF_KLT);

    // 10) alphas transpose, beta logits GEMM (M=2500, N=30000, K=300), softmax
    transpose_alpha<<<(Tt * Kk * Ee + 255) / 256, 256, 0, stream>>>(mu_q_a, ws + WS_ALPH);
    wmma_gemm_atb<<<GEMM_GRID(Tt * Kk, Vv), 32, 0, stream>>>(
        ws + WS_ALPH, emb, nullptr, out + OFF_BETA, Tt * Kk, Vv, Ee, 0);
    softmax_rows<<<Tt * Kk, 256, 0, stream>>>(out + OFF_BETA, Vv);

    // 11) kl_alpha partials + reduce
    kl_alpha_part<<<Tt * Kk, 256, 0, stream>>>(mu_q_a, ls_q_a, ws + WS_KLA);
    reduce_sum<<<1, 256, 0, stream>>>(ws + WS_KLA, Tt * Kk, 1.0f, out + OFF_KLA);

    // 12) px / lnpx / nll
    px_kernel<<<dim3(Bb, NVB), 256, 0, stream>>>(out + OFF_THETA, out + OFF_BETA,
                                                 X, ts, out + OFF_LNPX, ws + WS_NLL);
    reduce_sum<<<1, 256, 0, stream>>>(ws + WS_NLL, Bb * NVB, 1.0f / (float)Bb,
                                      out + OFF_NLL);
#undef GEMM_GRID
}